// GCNComm_68478958568087
// MI455X (gfx1250) — compile-verified
//
#include <hip/hip_runtime.h>
#include <hip/hip_bf16.h>
#include <stdint.h>

// GCN (PyG GCNConv semantics) x3 layers on MI455X (gfx1250, wave32).
//  - deg^{-1/2} symmetric norm, self-loops folded into agg init (no atomics).
//  - GEMM: V_WMMA_F32_16X16X4_F32, exact fp32, 32 WMMAs per 16x16 tile.
//  - Edge scatter: one wave32 per edge, float4 per lane, global_atomic_add_f32.

typedef float v2f __attribute__((ext_vector_type(2)));
typedef float v8f __attribute__((ext_vector_type(8)));

#define D 128  // feature dim for all layers

// ---------------- degree / normalization ----------------

__global__ __launch_bounds__(256) void deg_init_k(float* __restrict__ deg, int N) {
    int i = blockIdx.x * 256 + threadIdx.x;
    if (i < N) deg[i] = 1.0f;  // self-loop contribution
}

__global__ __launch_bounds__(256) void deg_accum_k(const long long* __restrict__ dst,
                                                   float* __restrict__ deg, int E) {
    int e = blockIdx.x * 256 + threadIdx.x;
    if (e < E) unsafeAtomicAdd(&deg[(int)dst[e]], 1.0f);
}

__global__ __launch_bounds__(256) void deg_rsqrt_k(float* __restrict__ deg, int N) {
    int i = blockIdx.x * 256 + threadIdx.x;
    if (i < N) {
        float d = deg[i];
        deg[i] = (d > 0.0f) ? rsqrtf(d) : 0.0f;  // deg>=1 here, guard matches ref
    }
}

// ---------------- fp32 WMMA GEMM: T = A(N x 128) @ W(128 x 128) ----------------
// One wave per 16x16 output tile. K-loop of 32 x V_WMMA_F32_16X16X4_F32.
// Fragment layouts per CDNA5 ISA 7.12.2:
//   A 16x4 f32 : lanes 0-15 -> M=lane, v0=K0,v1=K1 ; lanes 16-31 -> v0=K2,v1=K3
//   B 4x16 f32 : lanes 0-15 -> N=lane, v0=K0,v1=K1 ; lanes 16-31 -> v0=K2,v1=K3
//   C/D 16x16  : vj -> M=j (lanes 0-15) / M=j+8 (lanes 16-31), N=lane&15
__global__ __launch_bounds__(256) void gemm128_wmma_k(const float* __restrict__ A,
                                                      const float* __restrict__ W,
                                                      float* __restrict__ T, int N) {
    const int lane  = threadIdx.x & 31;
    const int w     = blockIdx.x * 8 + (threadIdx.x >> 5);  // global wave id
    const int rowT  = w >> 3;   // 8 column tiles (128/16)
    const int colT  = w & 7;
    if (rowT * 16 >= N) return;                 // wave-uniform: EXEC stays all-ones
    const int row0  = rowT * 16;
    const int col0  = colT * 16;
    const int half  = lane >> 4;                // 0 = lanes 0-15, 1 = lanes 16-31
    const int l     = lane & 15;

    v8f acc = {};
#pragma unroll
    for (int k = 0; k < D; k += 4) {
        const float* ap = A + (size_t)(row0 + l) * D + (k + half * 2);
        v2f a; a.x = ap[0]; a.y = ap[1];
        const float* bp = W + (size_t)(k + half * 2) * D + (col0 + l);
        v2f b; b.x = bp[0]; b.y = bp[D];
        acc = __builtin_amdgcn_wmma_f32_16x16x4_f32(
            /*neg_a=*/false, a, /*neg_b=*/false, b,
            /*c_mod=*/(short)0, acc, /*reuse_a=*/false, /*reuse_b=*/false);
    }

    float* out = T + (size_t)(row0 + half * 8) * D + (col0 + l);
#pragma unroll
    for (int j = 0; j < 8; ++j) out[(size_t)j * D] = acc[j];
}

// ---------------- aggregation ----------------

// agg[i,:] = t[i,:] * dinv[i]^2   (self-loop message; also zero-initializes agg)
__global__ __launch_bounds__(256) void agg_init_k(const float* __restrict__ T,
                                                  const float* __restrict__ dinv,
                                                  float* __restrict__ agg, int N) {
    int idx = blockIdx.x * 256 + threadIdx.x;
    if (idx < N * D) {
        float di = dinv[idx >> 7];
        agg[idx] = T[idx] * di * di;
    }
}

// one wave32 per edge: lane f handles features 4f..4f+3
__global__ __launch_bounds__(256) void edge_scatter_k(const long long* __restrict__ src,
                                                      const long long* __restrict__ dst,
                                                      const float* __restrict__ dinv,
                                                      const float* __restrict__ T,
                                                      float* __restrict__ agg, int E) {
    const int lane = threadIdx.x & 31;
    const int e = blockIdx.x * 8 + (threadIdx.x >> 5);
    if (e >= E) return;
    const int s = (int)src[e];
    const int d = (int)dst[e];
    const float wgt = dinv[s] * dinv[d];
    const float4 v = *(const float4*)(T + (size_t)s * D + lane * 4);
    float* o = agg + (size_t)d * D + lane * 4;
    unsafeAtomicAdd(o + 0, v.x * wgt);
    unsafeAtomicAdd(o + 1, v.y * wgt);
    unsafeAtomicAdd(o + 2, v.z * wgt);
    unsafeAtomicAdd(o + 3, v.w * wgt);
}

// out = relu(agg + b)
__global__ __launch_bounds__(256) void bias_relu_k(const float* __restrict__ agg,
                                                   const float* __restrict__ b,
                                                   float* __restrict__ out, int N) {
    int idx = blockIdx.x * 256 + threadIdx.x;
    if (idx < N * D) out[idx] = fmaxf(agg[idx] + b[idx & (D - 1)], 0.0f);
}

// ---------------- launch ----------------

extern "C" void kernel_launch(void* const* d_in, const int* in_sizes, int n_in,
                              void* d_out, int out_size, void* d_ws, size_t ws_size,
                              hipStream_t stream) {
    const float*     x   = (const float*)d_in[0];
    const long long* ei  = (const long long*)d_in[1];  // int64 per reference, [2,E]
    const float*     W1  = (const float*)d_in[2];
    const float*     b1  = (const float*)d_in[3];
    const float*     W2  = (const float*)d_in[4];
    const float*     b2  = (const float*)d_in[5];
    const float*     W3  = (const float*)d_in[6];
    const float*     b3  = (const float*)d_in[7];
    float*           out = (float*)d_out;

    const int N = in_sizes[0] / D;
    const int E = in_sizes[1] / 2;
    const long long* srcI = ei;
    const long long* dstI = ei + E;

    // workspace: dinv[N] | A[N*128] | B[N*128]   (~103 MB for N=100k)
    float* dinv = (float*)d_ws;
    size_t off = (((size_t)N * sizeof(float)) + 255) & ~(size_t)255;
    float* A = (float*)((char*)d_ws + off);
    float* B = A + (size_t)N * D;

    const int nBlk  = (N + 255) / 256;
    const int eBlk  = (E + 255) / 256;
    const int feBlk = (N * D + 255) / 256;       // 50k blocks for N=100k
    const int gBlk  = (N + 15) / 16;             // 8 waves/block = 8 col tiles
    const int sBlk  = (E + 7) / 8;               // 8 edges (waves) per block

    // symmetric deg^{-1/2} with self-loops
    deg_init_k <<<nBlk, 256, 0, stream>>>(dinv, N);
    deg_accum_k<<<eBlk, 256, 0, stream>>>(dstI, dinv, E);
    deg_rsqrt_k<<<nBlk, 256, 0, stream>>>(dinv, N);

    // layer 1: x -> B
    gemm128_wmma_k<<<gBlk, 256, 0, stream>>>(x, W1, A, N);
    agg_init_k    <<<feBlk, 256, 0, stream>>>(A, dinv, B, N);
    edge_scatter_k<<<sBlk, 256, 0, stream>>>(srcI, dstI, dinv, A, B, E);
    bias_relu_k   <<<feBlk, 256, 0, stream>>>(B, b1, B, N);

    // layer 2: B -> B
    gemm128_wmma_k<<<gBlk, 256, 0, stream>>>(B, W2, A, N);
    agg_init_k    <<<feBlk, 256, 0, stream>>>(A, dinv, B, N);
    edge_scatter_k<<<sBlk, 256, 0, stream>>>(srcI, dstI, dinv, A, B, E);
    bias_relu_k   <<<feBlk, 256, 0, stream>>>(B, b2, B, N);

    // layer 3: B -> out
    gemm128_wmma_k<<<gBlk, 256, 0, stream>>>(B, W3, A, N);
    agg_init_k    <<<feBlk, 256, 0, stream>>>(A, dinv, B, N);
    edge_scatter_k<<<sBlk, 256, 0, stream>>>(srcI, dstI, dinv, A, B, E);
    bias_relu_k   <<<feBlk, 256, 0, stream>>>(B, b3, out, N);
}